// GAT_ph_layer_55344948576543
// MI455X (gfx1250) — compile-verified
//
#include <hip/hip_runtime.h>

typedef __attribute__((ext_vector_type(2))) float v2f;
typedef __attribute__((ext_vector_type(8))) float v8f;

#define HDS 4      // heads
#define DIM 64     // features per head
#define HD  256    // HDS*DIM
#define NEG_SLOPE 0.2f

// ---------------- monotonic float<->uint key for atomic max ----------------
__device__ __forceinline__ unsigned f2key(float f) {
  unsigned u = __float_as_uint(f);
  return (u >> 31) ? ~u : (u | 0x80000000u);
}
__device__ __forceinline__ float key2f(unsigned k) {
  unsigned u = (k >> 31) ? (k & 0x7FFFFFFFu) : ~k;
  return __uint_as_float(u);
}

// ---------------- GEMM: h = x @ W  via V_WMMA_F32_16X16X4_F32 ----------------
// One wave per 16x16 tile of h[N,256]. K = 64 -> 16 WMMA ops.
// A (16x4 f32): lanes 0-15 hold K={0,1}, lanes 16-31 hold K={2,3} (2 VGPRs).
// B (4x16 f32): lanes 0-15 hold K={0,1} for N=lane, lanes 16-31 K={2,3}.
// C (16x16 f32): VGPR j: lanes 0-15 -> M=j, lanes 16-31 -> M=8+j.
__global__ __launch_bounds__(256)
void gemm_wmma_f32(const float* __restrict__ X, const float* __restrict__ W,
                   float* __restrict__ Hout, int n_rows) {
  const int wave = (int)((blockIdx.x * blockDim.x + threadIdx.x) >> 5);
  const int lane = threadIdx.x & 31;
  const int tiles_m = (n_rows + 15) >> 4;
  const int tm = wave >> 4;          // row tile   (16 col tiles: 256/16)
  const int tn = wave & 15;          // col tile
  if (tm >= tiles_m) return;
  const int row0 = tm << 4;
  const int col0 = tn << 4;
  const int hl  = lane >> 4;         // 0: lanes 0-15, 1: lanes 16-31
  const int l16 = lane & 15;
  const int rA  = min(row0 + l16, n_rows - 1);   // clamp (N=100000 is /16 anyway)

  v8f c = {};
#pragma unroll
  for (int kb = 0; kb < DIM; kb += 4) {
    const int ka = kb + hl * 2;
    v2f a, b;
    a.x = X[(size_t)rA * DIM + ka + 0];
    a.y = X[(size_t)rA * DIM + ka + 1];
    b.x = W[(size_t)(ka + 0) * HD + col0 + l16];
    b.y = W[(size_t)(ka + 1) * HD + col0 + l16];
    c = __builtin_amdgcn_wmma_f32_16x16x4_f32(false, a, false, b, (short)0, c,
                                              false, false);
  }
#pragma unroll
  for (int j = 0; j < 8; ++j) {
    const int row = row0 + hl * 8 + j;
    if (row < n_rows) Hout[(size_t)row * HD + col0 + l16] = c[j];
  }
}

// ---------------- el/er logits + init of m/s ----------------
__global__ __launch_bounds__(256)
void el_er_init(const float* __restrict__ Hf, const float* __restrict__ attn_l,
                const float* __restrict__ attn_r, float* __restrict__ el,
                float* __restrict__ er, unsigned* __restrict__ mkey,
                float* __restrict__ s, int nh_count) {
  const int idx = blockIdx.x * blockDim.x + threadIdx.x;  // n*HDS + h
  if (idx >= nh_count) return;
  const int n = idx >> 2, hh = idx & 3;
  const float4* hp = (const float4*)(Hf + (size_t)n * HD + hh * DIM);
  const float4* al = (const float4*)(attn_l + hh * DIM);
  const float4* ar = (const float4*)(attn_r + hh * DIM);
  float sl = 0.f, sr = 0.f;
#pragma unroll
  for (int i = 0; i < DIM / 4; ++i) {
    const float4 v = hp[i], a = al[i], b = ar[i];
    sl += v.x * a.x + v.y * a.y + v.z * a.z + v.w * a.w;
    sr += v.x * b.x + v.y * b.y + v.z * b.z + v.w * b.w;
  }
  el[idx] = sl;
  er[idx] = sr;
  mkey[idx] = 0u;    // == most-negative float in key space
  s[idx] = 0.f;
}

__global__ __launch_bounds__(256)
void zero_f32(float* __restrict__ p, int count) {
  const int i = blockIdx.x * blockDim.x + threadIdx.x;
  if (i < count) p[i] = 0.f;
}

// ---------------- edge pass 1: segment max ----------------
__global__ __launch_bounds__(256)
void edge_max(const int* __restrict__ src, const int* __restrict__ dst,
              const float* __restrict__ el, const float* __restrict__ er,
              unsigned* __restrict__ mkey, int e_count) {
  const int e = blockIdx.x * blockDim.x + threadIdx.x;
  if (e >= e_count) return;
  const int sN = src[e], dN = dst[e];
#pragma unroll
  for (int hh = 0; hh < HDS; ++hh) {
    float v = el[sN * HDS + hh] + er[dN * HDS + hh];
    v = v > 0.f ? v : NEG_SLOPE * v;
    atomicMax(&mkey[dN * HDS + hh], f2key(v));
  }
}

// ---------------- edge pass 2: segment sum of exp ----------------
__global__ __launch_bounds__(256)
void edge_sum(const int* __restrict__ src, const int* __restrict__ dst,
              const float* __restrict__ el, const float* __restrict__ er,
              const unsigned* __restrict__ mkey, float* __restrict__ s,
              int e_count) {
  const int e = blockIdx.x * blockDim.x + threadIdx.x;
  if (e >= e_count) return;
  const int sN = src[e], dN = dst[e];
#pragma unroll
  for (int hh = 0; hh < HDS; ++hh) {
    float v = el[sN * HDS + hh] + er[dN * HDS + hh];
    v = v > 0.f ? v : NEG_SLOPE * v;
    const float m = key2f(mkey[dN * HDS + hh]);
    atomicAdd(&s[dN * HDS + hh], __expf(v - m));
  }
}

// ---------------- edge pass 3: weighted aggregation (head-mean folded) ------
// One wave32 per edge; lane handles dims {2*lane, 2*lane+1} across all 4 heads.
__global__ __launch_bounds__(256)
void edge_aggregate(const int* __restrict__ src, const int* __restrict__ dst,
                    const float* __restrict__ el, const float* __restrict__ er,
                    const unsigned* __restrict__ mkey, const float* __restrict__ s,
                    const float* __restrict__ Hf, float* __restrict__ out_mean,
                    int e_count) {
  const int wave = (int)((blockIdx.x * blockDim.x + threadIdx.x) >> 5);
  const int lane = threadIdx.x & 31;
  if (wave >= e_count) return;
  const int sN = src[wave], dN = dst[wave];

  float a[HDS];
#pragma unroll
  for (int hh = 0; hh < HDS; ++hh) {
    float v = el[sN * HDS + hh] + er[dN * HDS + hh];
    v = v > 0.f ? v : NEG_SLOPE * v;
    const float m = key2f(mkey[dN * HDS + hh]);
    // fold 1/H head-mean into the coefficient: 4x less scatter traffic
    a[hh] = __expf(v - m) / s[dN * HDS + hh] * (1.0f / HDS);
  }

  float2 acc = make_float2(0.f, 0.f);
#pragma unroll
  for (int hh = 0; hh < HDS; ++hh) {
    const float2 v =
        *(const float2*)(Hf + (size_t)sN * HD + hh * DIM + lane * 2);
    acc.x += a[hh] * v.x;
    acc.y += a[hh] * v.y;
  }
  atomicAdd(&out_mean[(size_t)dN * DIM + lane * 2 + 0], acc.x);
  atomicAdd(&out_mean[(size_t)dN * DIM + lane * 2 + 1], acc.y);
}

// ---------------- residual mix ----------------
__global__ __launch_bounds__(256)
void residual_mix(const float* __restrict__ xin, const float* __restrict__ x0,
                  const float* __restrict__ om, const float* __restrict__ alpha_p,
                  const float* __restrict__ lamda_p, float* __restrict__ xout,
                  int count) {
  const int i = blockIdx.x * blockDim.x + threadIdx.x;
  if (i >= count) return;
  const float alpha = alpha_p[0];
  const float lamda = lamda_p[0];
  xout[i] = (1.0f - alpha) * xin[i] + alpha * om[i] + alpha * lamda * x0[i];
}

// ---------------- host launch ----------------
extern "C" void kernel_launch(void* const* d_in, const int* in_sizes, int n_in,
                              void* d_out, int out_size, void* d_ws, size_t ws_size,
                              hipStream_t stream) {
  const float* x      = (const float*)d_in[0];
  const float* x0     = (const float*)d_in[1];
  const int*   src    = (const int*)d_in[2];
  const int*   dst    = (const int*)d_in[3];
  const float* W      = (const float*)d_in[4];
  const float* attn_l = (const float*)d_in[5];
  const float* attn_r = (const float*)d_in[6];
  const float* alpha  = (const float*)d_in[7];
  const float* lamda  = (const float*)d_in[8];

  const int n = in_sizes[0] / DIM;   // 100000
  const int e = in_sizes[2];         // 1600000
  (void)n_in; (void)out_size; (void)ws_size;

  // workspace carve (all float32 unless noted)
  char* ws = (char*)d_ws;
  float*    h    = (float*)ws;    ws += (size_t)n * HD * 4;    // 102.4 MB
  float*    el   = (float*)ws;    ws += (size_t)n * HDS * 4;
  float*    er   = (float*)ws;    ws += (size_t)n * HDS * 4;
  unsigned* mkey = (unsigned*)ws; ws += (size_t)n * HDS * 4;
  float*    s    = (float*)ws;    ws += (size_t)n * HDS * 4;
  float*    om   = (float*)ws;    ws += (size_t)n * DIM * 4;   // 25.6 MB
  float*    xtmp = (float*)ws;                                 // 25.6 MB

  const int blk = 256;
  const int tiles_m = (n + 15) / 16;
  const int gemm_waves = tiles_m * (HD / 16);
  const dim3 g_gemm((gemm_waves * 32 + blk - 1) / blk);
  const dim3 g_nh((n * HDS + blk - 1) / blk);
  const dim3 g_nd((n * DIM + blk - 1) / blk);
  const dim3 g_e((e + blk - 1) / blk);
  const dim3 g_ew(((size_t)e * 32 + blk - 1) / blk);

  for (int step = 0; step < 2; ++step) {
    const float* xin = (step == 0) ? x : xtmp;
    float* xout = (step == 1) ? (float*)d_out : xtmp;

    gemm_wmma_f32<<<g_gemm, blk, 0, stream>>>(xin, W, h, n);
    el_er_init<<<g_nh, blk, 0, stream>>>(h, attn_l, attn_r, el, er, mkey, s,
                                         n * HDS);
    zero_f32<<<g_nd, blk, 0, stream>>>(om, n * DIM);
    edge_max<<<g_e, blk, 0, stream>>>(src, dst, el, er, mkey, e);
    edge_sum<<<g_e, blk, 0, stream>>>(src, dst, el, er, mkey, s, e);
    edge_aggregate<<<g_ew, blk, 0, stream>>>(src, dst, el, er, mkey, s, h, om, e);
    residual_mix<<<g_nd, blk, 0, stream>>>(xin, x0, om, alpha, lamda, xout,
                                           n * DIM);
  }
}